// SubspaceNet_9723805958224
// MI455X (gfx1250) — compile-verified
//
#include <hip/hip_runtime.h>
#include <math.h>

// ---------------------------------------------------------------------------
// SubspaceNet (MUSIC root finder), CDNA5/gfx1250.
// One wave32 per batch; 4 waves (128 threads) per block.
//  1) Hermitian-ize R into LDS
//  2) complex cyclic Jacobi eigensolve (A -> diag, V accumulates eigvecs)
//  3) noise-subspace weights w[k] from |lambda| descending rank
//  4) Fm = (V diag(w)) V^H via V_WMMA_F32_16X16X4_F32 (4 real GEMMs,
//     16 wmma issues / wave), accumulators in the native C/D layout
//  5) diagonal sums -> 31 complex coeffs; monic normalize
//  6) Durand-Kerner (40 iters) -> 30 roots; 2 Newton polish steps
//  7) doa_all = asin(angle/pi); rank by |(|z|-1)| (+1e6 if |z|>=1), first M
// ---------------------------------------------------------------------------

typedef float v2f __attribute__((ext_vector_type(2)));
typedef float v8f __attribute__((ext_vector_type(8)));

#define NDIM     16
#define DEG      30
#define NC       31
#define SWEEPS   8
#define DK_ITERS 40
#define NSTEPS   2

__global__ __launch_bounds__(128) void subspace_music_kernel(
    const float* __restrict__ Rzr, const float* __restrict__ Rzi,
    const int* __restrict__ Mp, float* __restrict__ out, int B)
{
  __shared__ float sAr[4][16][17], sAi[4][16][17];
  __shared__ float sVr[4][16][17], sVi[4][16][17];
  __shared__ float sCr[4][32],     sCi[4][32];
  __shared__ float sZr[4][32],     sZi[4][32];
  __shared__ float sW [4][16],     sKey[4][32];

  const int tid = threadIdx.x;
  const int wv  = tid >> 5;          // wave id in block
  const int l   = tid & 31;          // lane id
  const int b   = blockIdx.x * 4 + wv;
  if (b >= B) return;
  const int M   = Mp[0];

  float (*Ar)[17] = sAr[wv]; float (*Ai)[17] = sAi[wv];
  float (*Vr)[17] = sVr[wv]; float (*Vi)[17] = sVi[wv];
  float *Cr = sCr[wv], *Ci = sCi[wv];
  float *Zr = sZr[wv], *Zi = sZi[wv];
  float *W  = sW[wv],  *Key = sKey[wv];

  // ---- 1) load + Hermitian-ize:  R = 0.5*(Rz + Rz^H) --------------------
  const float* br = Rzr + (size_t)b * 256;
  const float* bi = Rzi + (size_t)b * 256;
  #pragma unroll
  for (int i = 0; i < 8; ++i) {
    int e = l + 32 * i;              // 0..255
    int m = e >> 4, n = e & 15;
    Ar[m][n] = 0.5f * (br[m * 16 + n] + br[n * 16 + m]);
    Ai[m][n] = 0.5f * (bi[m * 16 + n] - bi[n * 16 + m]);
    Vr[m][n] = (m == n) ? 1.0f : 0.0f;
    Vi[m][n] = 0.0f;
  }
  __syncthreads();

  // ---- 2) complex cyclic Jacobi ----------------------------------------
  for (int sweep = 0; sweep < SWEEPS; ++sweep) {
    for (int p = 0; p < NDIM - 1; ++p) {
      for (int q = p + 1; q < NDIM; ++q) {
        float app = Ar[p][p], aqq = Ar[q][q];
        float apr = Ar[p][q], api = Ai[p][q];
        float bmag = sqrtf(apr * apr + api * api);
        float tau  = (aqq - app) / (2.0f * bmag + 1e-30f);
        float t    = copysignf(1.0f, tau) / (fabsf(tau) + sqrtf(1.0f + tau * tau));
        float cc   = rsqrtf(1.0f + t * t);
        float ss   = t * cc;
        float er   = apr / (bmag + 1e-30f);
        float ei   = api / (bmag + 1e-30f);
        bool rot   = bmag > 1e-9f;
        float c  = rot ? cc : 1.0f;
        float sr = rot ? ss * er : 0.0f;
        float si = rot ? ss * ei : 0.0f;

        // column update: lanes 0-15 -> A, lanes 16-31 -> V
        int k = l & 15;
        if (l < 16) {
          float xr = Ar[k][p], xi = Ai[k][p];
          float yr = Ar[k][q], yi = Ai[k][q];
          Ar[k][p] = c * xr - (sr * yr + si * yi);
          Ai[k][p] = c * xi - (sr * yi - si * yr);
          Ar[k][q] = (sr * xr - si * xi) + c * yr;
          Ai[k][q] = (sr * xi + si * xr) + c * yi;
        } else {
          float xr = Vr[k][p], xi = Vi[k][p];
          float yr = Vr[k][q], yi = Vi[k][q];
          Vr[k][p] = c * xr - (sr * yr + si * yi);
          Vi[k][p] = c * xi - (sr * yi - si * yr);
          Vr[k][q] = (sr * xr - si * xi) + c * yr;
          Vi[k][q] = (sr * xi + si * xr) + c * yi;
        }
        __syncthreads();

        // row update on A: lanes 0-15, column j = lane
        if (l < 16) {
          int j = l;
          float xr = Ar[p][j], xi = Ai[p][j];
          float yr = Ar[q][j], yi = Ai[q][j];
          Ar[p][j] = c * xr - (sr * yr - si * yi);
          Ai[p][j] = c * xi - (sr * yi + si * yr);
          Ar[q][j] = (sr * xr + si * xi) + c * yr;
          Ai[q][j] = (sr * xi - si * xr) + c * yi;
        }
        __syncthreads();
      }
    }
  }

  // ---- 3) rank eigenvalues by |lambda| descending -> noise weights ------
  if (l < 16) Key[l] = fabsf(Ar[l][l]);
  __syncthreads();
  if (l < 16) {
    float mine = Key[l];
    int rank = 0;
    #pragma unroll
    for (int j = 0; j < 16; ++j) {
      float o = Key[j];
      rank += (o > mine) || (o == mine && j < l);
    }
    W[l] = (rank >= M) ? 1.0f : 0.0f;
  }
  __syncthreads();

  // ---- 4) Fm = (V diag(w)) V^H  via V_WMMA_F32_16X16X4_F32 --------------
  // A-operand (16x4): lane m = l&15, VGPR0 -> K = kb, VGPR1 -> K = kb+1,
  // kb = 4*kc + 2*(l>=16).  B-operand (4x16) of X^T has identical indexing.
  const int mm = l & 15;
  const int khalf = (l >> 4) << 1;

  v8f fr = {};                // Fm_real = Vr W Vr^T + Vi W Vi^T
  v8f g1 = {};                // Vi W Vr^T
  v8f g2 = {};                // Vr W Vi^T
  #pragma unroll
  for (int kc = 0; kc < 4; ++kc) {
    int kb = kc * 4 + khalf;
    float w0 = W[kb], w1 = W[kb + 1];
    v2f ar; ar.x = Vr[mm][kb] * w0; ar.y = Vr[mm][kb + 1] * w1;
    v2f ai; ai.x = Vi[mm][kb] * w0; ai.y = Vi[mm][kb + 1] * w1;
    v2f brv; brv.x = Vr[mm][kb];  brv.y = Vr[mm][kb + 1];
    v2f biv; biv.x = Vi[mm][kb];  biv.y = Vi[mm][kb + 1];
    fr = __builtin_amdgcn_wmma_f32_16x16x4_f32(false, ar, false, brv, (short)0, fr, false, false);
    fr = __builtin_amdgcn_wmma_f32_16x16x4_f32(false, ai, false, biv, (short)0, fr, false, false);
    g1 = __builtin_amdgcn_wmma_f32_16x16x4_f32(false, ai, false, brv, (short)0, g1, false, false);
    g2 = __builtin_amdgcn_wmma_f32_16x16x4_f32(false, ar, false, biv, (short)0, g2, false, false);
  }
  v8f fi = g1 - g2;           // Fm_imag

  __syncthreads();
  // scatter Fm back to LDS (reuse A): C/D layout M = v + 8*(l>=16), N = l&15
  #pragma unroll
  for (int v = 0; v < 8; ++v) {
    int mrow = v + 8 * (l >> 4);
    Ar[mrow][mm] = fr[v];
    Ai[mrow][mm] = fi[v];
  }
  __syncthreads();

  // ---- 5) coefficients: c[j] = trace(Fm, offset = 15 - j), j = 0..30 ----
  if (l < NC) {
    int off = 15 - l;
    int i0 = off < 0 ? -off : 0;
    int i1 = off < 0 ? 15 : 15 - off;
    float cr = 0.0f, ci = 0.0f;
    for (int i = i0; i <= i1; ++i) { cr += Ar[i][i + off]; ci += Ai[i][i + off]; }
    Cr[l] = cr; Ci[l] = ci;
  }
  __syncthreads();

  // monic normalize: c /= c[0]
  {
    float c0r = Cr[0], c0i = Ci[0];
    float myr = (l < NC) ? Cr[l] : 0.0f;
    float myi = (l < NC) ? Ci[l] : 0.0f;
    float d = c0r * c0r + c0i * c0i + 1e-30f;
    __syncthreads();
    if (l < NC) {
      Cr[l] = (myr * c0r + myi * c0i) / d;
      Ci[l] = (myi * c0r - myr * c0i) / d;
    }
  }
  __syncthreads();

  // ---- 6) Durand-Kerner + Newton polish ---------------------------------
  if (l < DEG) {
    float ang = 6.2831853071795864f * (float)l / (float)DEG + 0.35f;
    Zr[l] = 0.95f * __cosf(ang);
    Zi[l] = 0.95f * __sinf(ang);
  }
  __syncthreads();

  const int jj = (l < DEG) ? l : 0;
  float zr = 0.0f, zi = 0.0f;
  for (int it = 0; it < DK_ITERS; ++it) {
    zr = Zr[jj]; zi = Zi[jj];
    // p(z), monic coeffs (Cr[0] == 1)
    float pr = 0.0f, pi = 0.0f;
    for (int k = 0; k < NC; ++k) {
      float tr = pr * zr - pi * zi + Cr[k];
      pi = pr * zi + pi * zr + Ci[k];
      pr = tr;
    }
    // prod_{k != j} (z - z_k)
    float dr = 1.0f, di = 0.0f;
    for (int k = 0; k < DEG; ++k) {
      if (k == jj) continue;
      float dxr = zr - Zr[k], dxi = zi - Zi[k];
      float tr = dr * dxr - di * dxi;
      di = dr * dxi + di * dxr;
      dr = tr;
    }
    float den = dr * dr + di * di + 1e-30f;
    float qr = (pr * dr + pi * di) / den;
    float qi = (pi * dr - pr * di) / den;
    float nzr = zr - qr, nzi = zi - qi;
    __syncthreads();
    if (l < DEG) { Zr[l] = nzr; Zi[l] = nzi; }
    __syncthreads();
  }

  zr = Zr[jj]; zi = Zi[jj];
  for (int s = 0; s < NSTEPS; ++s) {
    float pr = 0.0f, pi = 0.0f;
    for (int k = 0; k < NC; ++k) {
      float tr = pr * zr - pi * zi + Cr[k];
      pi = pr * zi + pi * zr + Ci[k];
      pr = tr;
    }
    float dpr = 0.0f, dpi = 0.0f;        // p'(z): coeffs c[k]*(30-k), k<30
    for (int k = 0; k < DEG; ++k) {
      float sc = (float)(DEG - k);
      float tr = dpr * zr - dpi * zi + Cr[k] * sc;
      dpi = dpr * zi + dpi * zr + Ci[k] * sc;
      dpr = tr;
    }
    float den = dpr * dpr + dpi * dpi + 1e-30f;
    float qr = (pr * dpr + pi * dpi) / den;
    float qi = (pi * dpr - pr * dpi) / den;
    zr -= qr; zi -= qi;
  }

  // ---- 7) outputs --------------------------------------------------------
  const float inv_pi = 0.31830988618379067f;   // 1/(2*pi*DIST*FREQ), DIST=.5 FREQ=1
  float doa_val = 0.0f, key = 0.0f;
  if (l < DEG) {
    float ang = atan2f(zi, zr);
    doa_val = asinf(ang * inv_pi);
    out[(size_t)B * M + (size_t)b * DEG + l] = doa_val;   // doa_all
    float mag = sqrtf(zr * zr + zi * zi);
    key = fabsf(mag - 1.0f) + ((mag >= 1.0f) ? 1e6f : 0.0f);
    Key[l] = key;
  }
  __syncthreads();
  if (l < DEG) {
    int rank = 0;
    for (int k = 0; k < DEG; ++k) {
      float o = Key[k];
      rank += (o < key) || (o == key && k < l);
    }
    if (rank < M) out[(size_t)b * M + rank] = doa_val;    // doa (selected)
  }
}

extern "C" void kernel_launch(void* const* d_in, const int* in_sizes, int n_in,
                              void* d_out, int out_size, void* d_ws, size_t ws_size,
                              hipStream_t stream) {
  const float* Rzr = (const float*)d_in[0];
  const float* Rzi = (const float*)d_in[1];
  const int*   Mp  = (const int*)d_in[2];
  float* out = (float*)d_out;
  int B = in_sizes[0] / 256;             // B x 16 x 16
  dim3 grid((B + 3) / 4), block(128);
  subspace_music_kernel<<<grid, block, 0, stream>>>(Rzr, Rzi, Mp, out, B);
}